// CTCLoss_77902116814953
// MI455X (gfx1250) — compile-verified
//
#include <hip/hip_runtime.h>
#include <hip/hip_bf16.h>
#include <math.h>

#define T_  1024
#define B_  64
#define V_  512
#define L_  256
#define S_  (2 * L_ + 1)      // 513
#define NTHREADS 512
#define ROWS 8                // timesteps per TDM tile
#define NCHUNK (T_ / ROWS)    // 128
#define NEG_INF (-1e30f)

#define AS3 __attribute__((address_space(3)))

typedef unsigned int v4u __attribute__((ext_vector_type(4)));
typedef unsigned int v8u __attribute__((ext_vector_type(8)));

// ---- Tensor Data Mover: one instruction DMAs an 8x512 f32 tile (16KB) to LDS ----
// D# group0: [1:0]=count=1 | lds_addr | global_addr[56:0] | type=2
// D# group1: data_size=4B, tensor_dim0=V, tensor_dim1=T, tile_dim0=V, tile_dim1=ROWS,
//            tensor_dim0_stride=B*V (elements between consecutive t for fixed b)
__device__ __forceinline__ void tdm_load_tile(const float* gsrc, float* ldst) {
    unsigned long long ga = (unsigned long long)(size_t)gsrc;
    unsigned ldsoff = (unsigned)(size_t)(AS3 float*)ldst;

    v4u g0;
    g0[0] = 1u;                                               // count=1, user mode
    g0[1] = ldsoff;                                           // lds_addr (bytes)
    g0[2] = (unsigned)(ga & 0xFFFFFFFFu);                     // global_addr[31:0]
    g0[3] = (unsigned)((ga >> 32) & 0x1FFFFFFu) | (2u << 30); // global_addr[56:32] | type=2

    v8u g1;
    g1[0] = 2u << 16;                                         // workgroup_mask=0 | data_size=2 (4B)
    g1[1] = ((unsigned)V_ & 0xFFFFu) << 16;                   // tensor_dim0[15:0]
    g1[2] = (((unsigned)V_ >> 16) & 0xFFFFu)
          | (((unsigned)T_ & 0xFFFFu) << 16);                 // tensor_dim0[31:16] | tensor_dim1[15:0]
    g1[3] = (((unsigned)T_ >> 16) & 0xFFFFu)
          | (((unsigned)V_ & 0xFFFFu) << 16);                 // tensor_dim1[31:16] | tile_dim0
    g1[4] = (unsigned)ROWS;                                   // tile_dim1 | tile_dim2=0
    g1[5] = (unsigned)(B_ * V_);                              // tensor_dim0_stride[31:0]
    g1[6] = 0u;                                               // stride hi | dim1_stride lo
    g1[7] = 0u;                                               // dim1_stride hi

    asm volatile("tensor_load_to_lds %0, %1" :: "s"(g0), "s"(g1) : "memory");
}

__device__ __forceinline__ float wave_max(float x) {
#pragma unroll
    for (int d = 16; d >= 1; d >>= 1) x = fmaxf(x, __shfl_xor(x, d, 32));
    return x;
}
__device__ __forceinline__ float wave_sum(float x) {
#pragma unroll
    for (int d = 16; d >= 1; d >>= 1) x += __shfl_xor(x, d, 32);
    return x;
}
// Reduce 16 LDS partials redundantly in every wave (no extra barrier/broadcast).
__device__ __forceinline__ float xwave_max16(const float* red, int lane) {
    float v = red[lane & 15];
#pragma unroll
    for (int d = 8; d >= 1; d >>= 1) v = fmaxf(v, __shfl_xor(v, d, 32));
    return v;
}
__device__ __forceinline__ float xwave_sum16(const float* red, int lane) {
    float v = red[lane & 15];
#pragma unroll
    for (int d = 8; d >= 1; d >>= 1) v += __shfl_xor(v, d, 32);
    return v;
}

__global__ void ctc_zero_out(float* out) { out[0] = 0.0f; }

__global__ __launch_bounds__(NTHREADS)
void ctc_forward_kernel(const float* __restrict__ acts,
                        const int*   __restrict__ labels,
                        const int*   __restrict__ act_lens,
                        const int*   __restrict__ label_lens,
                        float*       __restrict__ out) {
    __shared__ __align__(16) float tile[2][ROWS][V_];  // double-buffered TDM target (32KB)
    __shared__ float alpha[2][S_ + 2];                 // ping-pong alpha, NEG_INF front pad
    __shared__ int   ext[S_];
    __shared__ int   skip[S_];
    __shared__ float redM[16];
    __shared__ float redS[16];
    __shared__ int   ibc;

    const int b    = blockIdx.x;
    const int tid  = threadIdx.x;
    const int lane = tid & 31;
    const int wid  = tid >> 5;

    const int llen = label_lens[b];
    const int alen = act_lens[b];

    if (tid == 0) {
        int off = 0;
        for (int i = 0; i < b; ++i) off += label_lens[i];
        ibc = off;
    }
    for (int s = tid; s < S_ + 2; s += NTHREADS) {
        alpha[0][s] = NEG_INF;
        alpha[1][s] = NEG_INF;
    }
    __syncthreads();
    const int off = ibc;

    for (int s = tid; s < S_; s += NTHREADS) {
        int lab = 0;
        if (s & 1) {
            int j = (s - 1) >> 1;
            int g = off + j;
            if (g < 0) g = 0;
            if (g > B_ * L_ - 1) g = B_ * L_ - 1;
            lab = (j < llen) ? labels[g] : 0;
        }
        ext[s] = lab;
    }
    __syncthreads();
    for (int s = tid; s < S_; s += NTHREADS) {
        int a = 0;
        if (s >= 2) {
            int e = ext[s];
            a = (e != 0 && e != ext[s - 2]) ? 1 : 0;
        }
        skip[s] = a;
    }

    // Wave 0 kicks off TDM DMA of chunk 0 (t = 0..7).
    if (tid < 32) tdm_load_tile(acts + (size_t)b * V_, &tile[0][0][0]);
    __syncthreads();  // publishes skip[]/ext[]

    int t = 0;
    for (int c = 0; c < NCHUNK; ++c) {
        const int cur = c & 1;
        const int nb  = cur ^ 1;

        if (tid < 32) {
            const int cp = (c + 1 < NCHUNK) ? (c + 1) : (NCHUNK - 1); // dummy refetch at tail
            tdm_load_tile(acts + ((size_t)cp * ROWS * B_ + b) * V_, &tile[nb][0][0]);
            __builtin_amdgcn_s_wait_tensorcnt(1);  // in-order: chunk c's tile has landed
        }
        __syncthreads();

        for (int r = 0; r < ROWS; ++r, ++t) {
            const float* rowp = tile[cur][r];

            // ---- logsumexp over 512 floats: 2 barriers ----
            const float x = rowp[tid];
            float m = wave_max(x);
            if (lane == 0) redM[wid] = m;
            __syncthreads();
            const float rmax = xwave_max16(redM, lane);

            float e = wave_sum(__expf(x - rmax));
            if (lane == 0) redS[wid] = e;
            __syncthreads();
            const float lse = rmax + __logf(xwave_sum16(redS, lane));

            // ---- alpha recurrence (thread tid -> state tid; thread 0 also state 512) ----
            const float* aOld = alpha[t & 1];
            float*       aNew = alpha[(t & 1) ^ 1];
#pragma unroll
            for (int k = 0; k < 2; ++k) {
                if (k == 1 && tid != 0) break;
                const int s = (k == 0) ? tid : (S_ - 1);
                const float lp = rowp[ext[s]] - lse;
                float nv;
                if (t == 0) {
                    nv = (s < 2) ? lp : NEG_INF;
                } else {
                    const float a1 = aOld[s + 2];
                    const float a2 = aOld[s + 1];
                    const float a3 = skip[s] ? aOld[s] : NEG_INF;
                    const float mm = fmaxf(a1, fmaxf(a2, a3));
                    const float sum = __expf(a1 - mm) + __expf(a2 - mm) + __expf(a3 - mm);
                    nv = lp + mm + __logf(sum);
                    if (t >= alen) nv = a1;  // reference: keep old alpha past act_len
                }
                aNew[s + 2] = nv;
            }
            __syncthreads();
        }
    }

    if (tid == 0) {
        const float* aF = alpha[T_ & 1];
        const int send = 2 * llen;
        const float a1 = aF[send + 2];
        int s2 = send - 1; if (s2 < 0) s2 = 0;
        const float a2 = aF[s2 + 2];
        const float mm = fmaxf(a1, a2);
        const float loss = -(mm + __logf(__expf(a1 - mm) + __expf(a2 - mm)));
        atomicAdd(out, loss);
    }
}

extern "C" void kernel_launch(void* const* d_in, const int* in_sizes, int n_in,
                              void* d_out, int out_size, void* d_ws, size_t ws_size,
                              hipStream_t stream) {
    const float* acts       = (const float*)d_in[0];
    const int*   labels     = (const int*)d_in[1];
    const int*   act_lens   = (const int*)d_in[2];
    const int*   label_lens = (const int*)d_in[3];
    float*       out        = (float*)d_out;

    ctc_zero_out<<<1, 1, 0, stream>>>(out);
    ctc_forward_kernel<<<B_, NTHREADS, 0, stream>>>(acts, labels, act_lens, label_lens, out);
}